// Memory_sup_57647051046931
// MI455X (gfx1250) — compile-verified
//
#include <hip/hip_runtime.h>
#include <hip/hip_bf16.h>
#include <math.h>

typedef __attribute__((ext_vector_type(2))) float v2f;
typedef __attribute__((ext_vector_type(8))) float v8f;

// D = A(16x4 f32) * B(4x16 f32) + C(16x16 f32)   -> v_wmma_f32_16x16x4_f32
__device__ __forceinline__ v8f wmma4(v2f a, v2f b, v8f c) {
  return __builtin_amdgcn_wmma_f32_16x16x4_f32(false, a, false, b, (short)0, c,
                                               false, false);
}

#define NPIX 65536   // B*H*W
#define CCH  256
#define MMEM 512
#define HW   4096
#define CHW  1048576 // C*H*W

// ---------------- pass 0: zero column-sum accumulator ----------------
__global__ void k_zero(float* colsum) {
  if (threadIdx.x < MMEM) colsum[threadIdx.x] = 0.0f;
}

// ---------------- pass 1a: per-pixel 1/||q||  (coalesced over pixels) ----
__global__ __launch_bounds__(256) void k_invnorm(const float* __restrict__ q,
                                                 float* __restrict__ invn) {
  const int b  = blockIdx.x >> 4;
  const int p  = (blockIdx.x & 15) * 256 + threadIdx.x;
  const float* base = q + b * CHW + p;
  float s = 0.0f;
  for (int c = 0; c < CCH; ++c) { float v = base[c * HW]; s += v * v; }
  invn[b * HW + p] = 1.0f / fmaxf(sqrtf(s), 1e-12f);
}

// ---------------- pass 1b: transpose BCHW -> q_flat[N,C], normalized ------
__global__ __launch_bounds__(256) void k_qflat(const float* __restrict__ q,
                                               const float* __restrict__ invn,
                                               float* __restrict__ qf) {
  __shared__ float t[32][33];
  const int b = blockIdx.z, ct = blockIdx.y, pt = blockIdx.x;
  const int tx = threadIdx.x, ty = threadIdx.y;
  const float* src = q + b * CHW + (ct * 32) * HW + pt * 32;
#pragma unroll
  for (int j = 0; j < 4; ++j)
    t[ty + j * 8][tx] = src[(ty + j * 8) * HW + tx];  // [c_local][p_local]
  __syncthreads();
  const int n0 = b * HW + pt * 32;
#pragma unroll
  for (int j = 0; j < 4; ++j) {
    const int p = ty + j * 8;
    qf[(n0 + p) * CCH + ct * 32 + tx] = t[tx][p] * invn[n0 + p];
  }
}

// ---------------- pass 2: E = exp(q_flat @ key^T - 1)  (WMMA fp32) --------
// each wave: one 16x64 strip; 4 accumulators reuse the A fragment 4x.
__global__ __launch_bounds__(128) void k_gemm1(const float* __restrict__ qf,
                                               const float* __restrict__ key,
                                               float* __restrict__ E) {
  const int lane = threadIdx.x & 31, wv = threadIdx.x >> 5;
  const int gid  = blockIdx.x * 4 + wv;        // 0..32767
  const int rowt = gid >> 3;                   // 0..4095
  const int mb   = (gid & 7) * 64;             // col strip base
  const int l16 = lane & 15, hl = lane >> 4;
  const int koff = hl * 2, rowh = hl * 8;
  const int row = rowt * 16 + l16;

  v8f a0 = {}, a1 = {}, a2 = {}, a3 = {};
  const float* ap = qf + row * CCH + koff;
  // B(k,col) = key[col*C + k]; VGPR pair (k+koff, k+koff+1) is contiguous.
  const float* b0 = key + (mb +  0 + l16) * CCH + koff;
  const float* b1 = key + (mb + 16 + l16) * CCH + koff;
  const float* b2 = key + (mb + 32 + l16) * CCH + koff;
  const float* b3 = key + (mb + 48 + l16) * CCH + koff;
  for (int k = 0; k < CCH; k += 4) {
    v2f a = *(const v2f*)(ap + k);
    a0 = wmma4(a, *(const v2f*)(b0 + k), a0);
    a1 = wmma4(a, *(const v2f*)(b1 + k), a1);
    a2 = wmma4(a, *(const v2f*)(b2 + k), a2);
    a3 = wmma4(a, *(const v2f*)(b3 + k), a3);
  }
  float* er = E + (rowt * 16 + rowh) * MMEM + mb + l16;
#pragma unroll
  for (int r = 0; r < 8; ++r) {
    er[r * MMEM +  0] = expf(a0[r] - 1.0f);
    er[r * MMEM + 16] = expf(a1[r] - 1.0f);
    er[r * MMEM + 32] = expf(a2[r] - 1.0f);
    er[r * MMEM + 48] = expf(a3[r] - 1.0f);
  }
}

// ------- pass 3: row softmax -> smm; accumulate column sums of E ----------
__global__ __launch_bounds__(256) void k_rowsoft(const float* __restrict__ E,
                                                 float* __restrict__ smm,
                                                 float* __restrict__ colsum) {
  __shared__ float cpart[MMEM];
  for (int i = threadIdx.x; i < MMEM; i += 256) cpart[i] = 0.0f;
  __syncthreads();
  const int wv = threadIdx.x >> 5, lane = threadIdx.x & 31;
  const int row0 = blockIdx.x * 64 + wv * 8;
  for (int r = 0; r < 8; ++r) {
    const float* er = E + (row0 + r) * MMEM + lane;
    float e[16];
    float s = 0.0f;
#pragma unroll
    for (int j = 0; j < 16; ++j) { e[j] = er[j * 32]; s += e[j]; }
#pragma unroll
    for (int off = 16; off; off >>= 1) s += __shfl_xor(s, off, 32);
    const float inv = 1.0f / s;
    float* orow = smm + (row0 + r) * MMEM + lane;
#pragma unroll
    for (int j = 0; j < 16; ++j) {
      orow[j * 32] = e[j] * inv;
      atomicAdd(&cpart[lane + j * 32], e[j]);
    }
  }
  __syncthreads();
  for (int i = threadIdx.x; i < MMEM; i += 256) atomicAdd(&colsum[i], cpart[i]);
}

// ------- pass 4: scale E in place by 1/colsum -> softmax over queries -----
__global__ __launch_bounds__(256) void k_colscale(float* __restrict__ E,
                                                  const float* __restrict__ colsum) {
  const int idx = blockIdx.x * 256 + threadIdx.x;  // float4 index
  const int m = (idx & 127) * 4;                    // 128 float4 per row
  float4* p = ((float4*)E) + idx;
  float4 v = *p;
  v.x /= colsum[m]; v.y /= colsum[m + 1]; v.z /= colsum[m + 2]; v.w /= colsum[m + 3];
  *p = v;
}

// ---------------- pass 5: concat = smm[N,512] @ key[512,256] (WMMA) -------
__global__ __launch_bounds__(128) void k_gemm2(const float* __restrict__ smm,
                                               const float* __restrict__ key,
                                               float* __restrict__ cm) {
  const int lane = threadIdx.x & 31, wv = threadIdx.x >> 5;
  const int gid  = blockIdx.x * 4 + wv;        // 0..16383
  const int rowt = gid >> 2;
  const int cb   = (gid & 3) * 64;
  const int l16 = lane & 15, hl = lane >> 4;
  const int koff = hl * 2, rowh = hl * 8;
  const int row = rowt * 16 + l16;

  v8f a0 = {}, a1 = {}, a2 = {}, a3 = {};
  const float* ap = smm + row * MMEM + koff;
  // B(k,col) = key[k*256 + col]: rows k+koff and k+koff+1, stride 256.
  const float* p0 = key + cb +  0 + l16;
  const float* p1 = key + cb + 16 + l16;
  const float* p2 = key + cb + 32 + l16;
  const float* p3 = key + cb + 48 + l16;
  for (int k = 0; k < MMEM; k += 4) {
    v2f a = *(const v2f*)(ap + k);
    const int r0 = (k + koff) * CCH, r1 = (k + koff + 1) * CCH;
    v2f b;
    b.x = p0[r0]; b.y = p0[r1]; a0 = wmma4(a, b, a0);
    b.x = p1[r0]; b.y = p1[r1]; a1 = wmma4(a, b, a1);
    b.x = p2[r0]; b.y = p2[r1]; a2 = wmma4(a, b, a2);
    b.x = p3[r0]; b.y = p3[r1]; a3 = wmma4(a, b, a3);
  }
  float* cr = cm + (rowt * 16 + rowh) * CCH + cb + l16;
#pragma unroll
  for (int r = 0; r < 8; ++r) {
    cr[r * CCH +  0] = a0[r];
    cr[r * CCH + 16] = a1[r];
    cr[r * CCH + 32] = a2[r];
    cr[r * CCH + 48] = a3[r];
  }
}

// -------- pass 6: y = [qf | cm] @ conv_w^T, written straight to BCHW ------
__global__ __launch_bounds__(128) void k_gemm3(const float* __restrict__ qf,
                                               const float* __restrict__ cm,
                                               const float* __restrict__ cw,
                                               float* __restrict__ y) {
  const int lane = threadIdx.x & 31, wv = threadIdx.x >> 5;
  const int gid  = blockIdx.x * 4 + wv;        // 0..16383
  const int rowt = gid >> 2;
  const int ob   = (gid & 3) * 64;
  const int l16 = lane & 15, hl = lane >> 4;
  const int koff = hl * 2, rowh = hl * 8;
  const int row = rowt * 16 + l16;

  v8f a0 = {}, a1 = {}, a2 = {}, a3 = {};
  // B(k,col) = cw[col*512 + k]; (k+koff, k+koff+1) contiguous.
  const float* b0 = cw + (ob +  0 + l16) * 512 + koff;
  const float* b1 = cw + (ob + 16 + l16) * 512 + koff;
  const float* b2 = cw + (ob + 32 + l16) * 512 + koff;
  const float* b3 = cw + (ob + 48 + l16) * 512 + koff;
  const float* ap = qf + row * CCH + koff;
  for (int k = 0; k < CCH; k += 4) {   // first half of K: q_flat
    v2f a = *(const v2f*)(ap + k);
    a0 = wmma4(a, *(const v2f*)(b0 + k), a0);
    a1 = wmma4(a, *(const v2f*)(b1 + k), a1);
    a2 = wmma4(a, *(const v2f*)(b2 + k), a2);
    a3 = wmma4(a, *(const v2f*)(b3 + k), a3);
  }
  ap = cm + row * CCH + koff;
  for (int k = 0; k < CCH; k += 4) {   // second half of K: concat_memory
    v2f a = *(const v2f*)(ap + k);
    a0 = wmma4(a, *(const v2f*)(b0 + 256 + k), a0);
    a1 = wmma4(a, *(const v2f*)(b1 + 256 + k), a1);
    a2 = wmma4(a, *(const v2f*)(b2 + 256 + k), a2);
    a3 = wmma4(a, *(const v2f*)(b3 + 256 + k), a3);
  }
  // scatter into BCHW: n = rowt*16 + rowh + r ; out[b, o, n%4096]
  const int n0 = rowt * 16 + rowh;
  const int bb = n0 >> 12, p0 = n0 & (HW - 1);
  float* y0 = y + bb * CHW + (ob +  0 + l16) * HW + p0;
  float* y1 = y + bb * CHW + (ob + 16 + l16) * HW + p0;
  float* y2 = y + bb * CHW + (ob + 32 + l16) * HW + p0;
  float* y3 = y + bb * CHW + (ob + 48 + l16) * HW + p0;
#pragma unroll
  for (int r = 0; r < 8; ++r) {
    y0[r] = a0[r]; y1[r] = a1[r]; y2[r] = a2[r]; y3[r] = a3[r];
  }
}

// -------- pass 7: per-channel batch stats -> scale/shift ------------------
__global__ __launch_bounds__(256) void k_chanstat(const float* __restrict__ y,
                                                  const float* __restrict__ g,
                                                  const float* __restrict__ be,
                                                  float* __restrict__ scale,
                                                  float* __restrict__ shift) {
  __shared__ float ss[256], sq[256];
  const int o = blockIdx.x;
  float s = 0.0f, q2 = 0.0f;
  for (int b = 0; b < 16; ++b) {
    const float* base = y + b * CHW + o * HW + threadIdx.x;
#pragma unroll
    for (int j = 0; j < 16; ++j) { float v = base[j * 256]; s += v; q2 += v * v; }
  }
  ss[threadIdx.x] = s; sq[threadIdx.x] = q2;
  __syncthreads();
  for (int st = 128; st; st >>= 1) {
    if (threadIdx.x < st) {
      ss[threadIdx.x] += ss[threadIdx.x + st];
      sq[threadIdx.x] += sq[threadIdx.x + st];
    }
    __syncthreads();
  }
  if (threadIdx.x == 0) {
    const float mean = ss[0] * (1.0f / 65536.0f);
    const float var  = sq[0] * (1.0f / 65536.0f) - mean * mean; // biased
    const float sc = g[o] * rsqrtf(var + 1e-5f);
    scale[o] = sc;
    shift[o] = be[o] - mean * sc;
  }
}

// -------- pass 8: BN + ReLU in place on BCHW ------------------------------
__global__ __launch_bounds__(256) void k_bnrelu(float* __restrict__ y,
                                                const float* __restrict__ scale,
                                                const float* __restrict__ shift) {
  const int idx = blockIdx.x * 256 + threadIdx.x;  // float4 index
  const int o = (idx >> 10) & 255;                  // (idx*4 / 4096) % 256
  float4* p = ((float4*)y) + idx;
  float4 v = *p;
  const float sc = scale[o], sh = shift[o];
  v.x = fmaxf(v.x * sc + sh, 0.0f);
  v.y = fmaxf(v.y * sc + sh, 0.0f);
  v.z = fmaxf(v.z * sc + sh, 0.0f);
  v.w = fmaxf(v.w * sc + sh, 0.0f);
  *p = v;
}

extern "C" void kernel_launch(void* const* d_in, const int* in_sizes, int n_in,
                              void* d_out, int out_size, void* d_ws, size_t ws_size,
                              hipStream_t stream) {
  const float* query = (const float*)d_in[0];
  const float* key   = (const float*)d_in[1];
  const float* cw    = (const float*)d_in[2];
  const float* gamma = (const float*)d_in[3];
  const float* beta  = (const float*)d_in[4];
  (void)in_sizes; (void)n_in; (void)out_size; (void)ws_size;

  float* out = (float*)d_out;
  float* yq  = out;               // updated_query [16,256,64,64]
  float* smq = out + 16777216;    // softmax_score_query [N,M] (E, then scaled)
  float* smm = out + 50331648;    // softmax_score_memory [N,M]

  float* ws     = (float*)d_ws;
  float* qf     = ws;                // [N,256]   64 MB
  float* cm     = ws + 16777216;     // [N,256]   64 MB
  float* invn   = ws + 33554432;     // [N]
  float* colsum = ws + 33619968;     // [512]
  float* scale  = ws + 33620480;     // [256]
  float* shift  = ws + 33620736;     // [256]

  k_zero    <<<1, 512, 0, stream>>>(colsum);
  k_invnorm <<<256, 256, 0, stream>>>(query, invn);
  k_qflat   <<<dim3(128, 8, 16), dim3(32, 8), 0, stream>>>(query, invn, qf);
  k_gemm1   <<<8192, 128, 0, stream>>>(qf, key, smq);        // E = exp(score-1)
  k_rowsoft <<<1024, 256, 0, stream>>>(smq, smm, colsum);
  k_colscale<<<32768, 256, 0, stream>>>(smq, colsum);
  k_gemm2   <<<4096, 128, 0, stream>>>(smm, key, cm);
  k_gemm3   <<<4096, 128, 0, stream>>>(qf, cm, cw, yq);
  k_chanstat<<<256, 256, 0, stream>>>(yq, gamma, beta, scale, shift);
  k_bnrelu  <<<16384, 256, 0, stream>>>(yq, scale, shift);
}